// Mamba2Blocks_22282290332048
// MI455X (gfx1250) — compile-verified
//
#include <hip/hip_runtime.h>
#include <hip/hip_bf16.h>
#include <math.h>

#define D_MODEL   256
#define NB        2
#define D_STATE   64
#define D_CONV    4
#define D_INNER   512
#define HEADDIM   64
#define NHEADS    8
#define CONV_DIM  640          // D_INNER + 2*D_STATE
#define D_IN_PROJ 1160         // 2*D_INNER + 2*D_STATE + NHEADS
#define BATCH     8
#define SEQLEN    1024
#define MROWS     (BATCH * SEQLEN)   // 8192
#define EPS       1e-5f

typedef __attribute__((ext_vector_type(2))) float v2f;
typedef __attribute__((ext_vector_type(8))) float v8f;

// LDS row stride: 36 floats = 144 B -> 16B-aligned rows (B128-async friendly)
// and 36*row mod 64 hits 16 distinct banks over 16 rows (conflict-free frags).
#define LSTRIDE 36

__device__ __forceinline__ void async_ld_b128(unsigned lds_addr, const float* gptr) {
    asm volatile("global_load_async_to_lds_b128 %0, %1, off"
                 :: "v"(lds_addr), "v"((unsigned long long)(const void*)gptr)
                 : "memory");
}
__device__ __forceinline__ void wait_async0() {
    asm volatile("s_wait_asynccnt 0x0" ::: "memory");
}

// ---------------------------------------------------------------------------
// fp32 WMMA GEMM:  C[m,n] = sum_k A[m*K+k] * W[n*K+k]
// Block tile 64(M) x 64(N), 8 waves (4x2); each wave owns a 16x32 strip
// (two 16x16 accumulators sharing one A fragment). K staged in LDS chunks
// of 32 via GLOBAL_LOAD_ASYNC_TO_LDS_B128 (ASYNCcnt), 16 WMMA per chunk.
// OOB rows are clamped (not zeroed): clamped garbage only reaches C rows/
// cols >= M/N, which the masked store never writes.
// ---------------------------------------------------------------------------
__global__ __launch_bounds__(256)
void gemm_xt_wmma_f32(const float* __restrict__ A,
                      const float* __restrict__ W,
                      float* __restrict__ C,
                      int M, int N, int K) {
    __shared__ float Asub[64][LSTRIDE];
    __shared__ float Wsub[64][LSTRIDE];

    const int tid  = threadIdx.x;
    const int lane = tid & 31;
    const int wave = tid >> 5;          // 0..7
    const int wm   = wave >> 1;         // 0..3  (M sub-tile, 16 rows)
    const int wn   = wave & 1;          // 0..1  (N strip, 32 cols)

    const int m0 = blockIdx.x * 64;
    const int n0 = blockIdx.y * 64;

    // staging indices: 64 rows x 32 K per tile, 8 floats (2 x b128) / thread
    const int lr = tid >> 2;            // 0..63
    const int lc = (tid & 3) * 8;       // 0,8,16,24

    const int arow = min(m0 + lr, M - 1);   // clamp (see header comment)
    const int wrow = min(n0 + lr, N - 1);

    const unsigned la = (unsigned)(unsigned long long)&Asub[lr][lc];
    const unsigned lw = (unsigned)(unsigned long long)&Wsub[lr][lc];

    v8f acc0 = {};
    v8f acc1 = {};

    const int frow  = lane & 15;             // fragment row/col within 16
    const int khalf = (lane >> 4) * 2;       // lanes16-31 -> K+2
    const int mhalf = (lane >> 4) * 8;       // C/D: lanes16-31 -> M+8

    for (int k0 = 0; k0 < K; k0 += 32) {
        const float* ga = A + (size_t)arow * K + k0 + lc;
        const float* gw = W + (size_t)wrow * K + k0 + lc;
        async_ld_b128(la,      ga);
        async_ld_b128(la + 16, ga + 4);
        async_ld_b128(lw,      gw);
        async_ld_b128(lw + 16, gw + 4);
        wait_async0();
        __syncthreads();

        #pragma unroll
        for (int k4 = 0; k4 < 32; k4 += 4) {
            v2f a, b0, b1;
            a.x  = Asub[wm * 16 + frow][k4 + khalf + 0];
            a.y  = Asub[wm * 16 + frow][k4 + khalf + 1];
            b0.x = Wsub[wn * 32 + frow][k4 + khalf + 0];
            b0.y = Wsub[wn * 32 + frow][k4 + khalf + 1];
            b1.x = Wsub[wn * 32 + 16 + frow][k4 + khalf + 0];
            b1.y = Wsub[wn * 32 + 16 + frow][k4 + khalf + 1];
            acc0 = __builtin_amdgcn_wmma_f32_16x16x4_f32(
                false, a, false, b0, (short)0, acc0, false, false);
            acc1 = __builtin_amdgcn_wmma_f32_16x16x4_f32(
                false, a, false, b1, (short)0, acc1, false, false);
        }
        __syncthreads();
    }

    // store: VGPR v -> M = v (+8 for hi lanes), N = lane
    const int gcol0 = n0 + wn * 32 + frow;
    const int gcol1 = gcol0 + 16;
    #pragma unroll
    for (int v = 0; v < 8; ++v) {
        const int grow = m0 + wm * 16 + mhalf + v;
        if (grow < M) {
            if (gcol0 < N) C[(size_t)grow * N + gcol0] = acc0[v];
            if (gcol1 < N) C[(size_t)grow * N + gcol1] = acc1[v];
        }
    }
}

// ---------------------------------------------------------------------------
// dt = softplus(zx[:, 1152+h] + dt_bias[h])
// ---------------------------------------------------------------------------
__global__ __launch_bounds__(256)
void dt_softplus_kernel(const float* __restrict__ zx,
                        const float* __restrict__ dt_bias,
                        float* __restrict__ dt_out) {
    int idx = blockIdx.x * 256 + threadIdx.x;           // MROWS*NHEADS
    if (idx >= MROWS * NHEADS) return;
    int h  = idx & (NHEADS - 1);
    int bl = idx >> 3;
    float v = zx[(size_t)bl * D_IN_PROJ + (D_IN_PROJ - NHEADS) + h] + dt_bias[h];
    dt_out[idx] = (v > 20.f) ? v : log1pf(expf(v));
}

// ---------------------------------------------------------------------------
// depthwise causal conv(4) + bias + SiLU over xBC slice of zx
// ---------------------------------------------------------------------------
__global__ __launch_bounds__(256)
void conv_silu_kernel(const float* __restrict__ zx,
                      const float* __restrict__ cw,
                      const float* __restrict__ cb,
                      float* __restrict__ xbc) {
    int idx = blockIdx.x * 256 + threadIdx.x;           // MROWS*CONV_DIM
    if (idx >= MROWS * CONV_DIM) return;
    int c  = idx % CONV_DIM;
    int bl = idx / CONV_DIM;
    int l  = bl & (SEQLEN - 1);
    int b  = bl >> 10;

    float acc = cb[c];
    #pragma unroll
    for (int k = 0; k < D_CONV; ++k) {
        int ls = l - (D_CONV - 1) + k;
        if (ls >= 0)
            acc += cw[c * D_CONV + k] *
                   zx[((size_t)(b * SEQLEN + ls)) * D_IN_PROJ + D_INNER + c];
    }
    xbc[(size_t)bl * CONV_DIM + c] = acc / (1.f + expf(-acc)); // silu
}

// ---------------------------------------------------------------------------
// Selective-state scan: one block per (b,h); 64x64 state in registers
// (16 f32 per thread). 1024 sequential steps; LDS broadcasts x/B/C and
// reduces the C.s contraction across the 4 state-column groups.
// ---------------------------------------------------------------------------
__global__ __launch_bounds__(256)
void scan_kernel(const float* __restrict__ xbc,
                 const float* __restrict__ dt_buf,
                 const float* __restrict__ A_log,
                 const float* __restrict__ Dvec,
                 float* __restrict__ ybuf) {
    __shared__ float lds_in[192];     // x[64] | B[64] | C[64]
    __shared__ float lds_acc[256];

    const int bid = blockIdx.x;               // 0..63
    const int b   = bid >> 3;
    const int h   = bid & (NHEADS - 1);
    const int t   = threadIdx.x;
    const int p   = t & 63;                   // headdim index
    const int sb  = (t >> 6) * 16;            // state-col base (0,16,32,48)

    const float Acoef = -expf(A_log[h]);
    const float Dh    = Dvec[h];

    float s[16];
    #pragma unroll
    for (int j = 0; j < 16; ++j) s[j] = 0.f;

    for (int l = 0; l < SEQLEN; ++l) {
        const size_t rowx = (size_t)(b * SEQLEN + l) * CONV_DIM;
        if (t < 64)       lds_in[t] = xbc[rowx + h * HEADDIM + t];
        else if (t < 128) lds_in[t] = xbc[rowx + D_INNER + (t - 64)];
        else if (t < 192) lds_in[t] = xbc[rowx + D_INNER + D_STATE + (t - 128)];
        __syncthreads();

        const float dt  = dt_buf[(size_t)(b * SEQLEN + l) * NHEADS + h];
        const float dA  = expf(dt * Acoef);
        const float xv  = lds_in[p];
        const float cfx = dt * xv;

        float acc = 0.f;
        #pragma unroll
        for (int j = 0; j < 16; ++j) {
            const float Bv = lds_in[64  + sb + j];
            const float Cv = lds_in[128 + sb + j];
            s[j] = s[j] * dA + cfx * Bv;
            acc += s[j] * Cv;
        }
        lds_acc[t] = acc;
        __syncthreads();

        if (t < 64) {
            float yv = lds_acc[t] + lds_acc[t + 64] +
                       lds_acc[t + 128] + lds_acc[t + 192] + Dh * xv;
            ybuf[(size_t)(b * SEQLEN + l) * D_INNER + h * HEADDIM + t] = yv;
        }
    }
}

// ---------------------------------------------------------------------------
// y = y * silu(z); y *= rsqrt(mean(y^2)+eps) * norm_w   (row = 512 elems)
// ---------------------------------------------------------------------------
__global__ __launch_bounds__(256)
void gate_rmsnorm_kernel(const float* __restrict__ zx,
                         const float* __restrict__ norm_w,
                         float* __restrict__ ybuf) {
    __shared__ float red[256];
    const int row = blockIdx.x;               // 0..MROWS-1
    const int t   = threadIdx.x;

    float v[2];
    #pragma unroll
    for (int i = 0; i < 2; ++i) {
        const int c = t + i * 256;
        const float z  = zx[(size_t)row * D_IN_PROJ + c];
        const float yv = ybuf[(size_t)row * D_INNER + c];
        v[i] = yv * (z / (1.f + expf(-z)));
    }
    red[t] = v[0] * v[0] + v[1] * v[1];
    __syncthreads();
    #pragma unroll
    for (int off = 128; off > 0; off >>= 1) {
        if (t < off) red[t] += red[t + off];
        __syncthreads();
    }
    const float scale = rsqrtf(red[0] / (float)D_INNER + EPS);
    #pragma unroll
    for (int i = 0; i < 2; ++i) {
        const int c = t + i * 256;
        ybuf[(size_t)row * D_INNER + c] = v[i] * scale * norm_w[c];
    }
}

// ---------------------------------------------------------------------------
extern "C" void kernel_launch(void* const* d_in, const int* in_sizes, int n_in,
                              void* d_out, int out_size, void* d_ws, size_t ws_size,
                              hipStream_t stream) {
    const float* x         = (const float*)d_in[0];
    const float* in_proj_w = (const float*)d_in[1];
    const float* conv_w    = (const float*)d_in[2];
    const float* conv_b    = (const float*)d_in[3];
    const float* dt_bias   = (const float*)d_in[4];
    const float* A_log     = (const float*)d_in[5];
    const float* Dvec      = (const float*)d_in[6];
    const float* norm_w    = (const float*)d_in[7];
    const float* out_proj  = (const float*)d_in[8];
    float* out = (float*)d_out;

    // workspace layout (floats)
    float* ws   = (float*)d_ws;
    float* hbuf = ws;                                   // 8192*256
    float* zx   = hbuf + (size_t)MROWS * D_MODEL;       // 8192*1160
    float* xbc  = zx   + (size_t)MROWS * D_IN_PROJ;     // 8192*640
    float* dtb  = xbc  + (size_t)MROWS * CONV_DIM;      // 8192*8
    float* ybuf = dtb  + (size_t)MROWS * NHEADS;        // 8192*512

    const dim3 blk(256);
    const dim3 g_in((MROWS + 63) / 64, (D_IN_PROJ + 63) / 64);  // 128 x 19
    const dim3 g_out((MROWS + 63) / 64, (D_MODEL  + 63) / 64);  // 128 x 4

    for (int i = 0; i < NB; ++i) {
        const float* wi = in_proj_w + (size_t)i * D_IN_PROJ * D_MODEL;
        const float* cw = conv_w    + (size_t)i * CONV_DIM * D_CONV;
        const float* cb = conv_b    + (size_t)i * CONV_DIM;
        const float* db = dt_bias   + (size_t)i * NHEADS;
        const float* al = A_log     + (size_t)i * NHEADS;
        const float* dv = Dvec      + (size_t)i * NHEADS;
        const float* nw = norm_w    + (size_t)i * D_INNER;
        const float* wo = out_proj  + (size_t)i * D_MODEL * D_INNER;

        const float* inA  = (i == 0) ? x : hbuf;
        float*       outC = (i == NB - 1) ? out : hbuf;

        gemm_xt_wmma_f32<<<g_in, blk, 0, stream>>>(inA, wi, zx,
                                                   MROWS, D_IN_PROJ, D_MODEL);
        dt_softplus_kernel<<<(MROWS * NHEADS + 255) / 256, blk, 0, stream>>>(zx, db, dtb);
        conv_silu_kernel<<<(MROWS * CONV_DIM + 255) / 256, blk, 0, stream>>>(zx, cw, cb, xbc);
        scan_kernel<<<BATCH * NHEADS, blk, 0, stream>>>(xbc, dtb, al, dv, ybuf);
        gate_rmsnorm_kernel<<<MROWS, blk, 0, stream>>>(zx, nw, ybuf);
        gemm_xt_wmma_f32<<<g_out, blk, 0, stream>>>(ybuf, wo, outC,
                                                    MROWS, D_MODEL, D_INNER);
    }
    (void)in_sizes; (void)n_in; (void)out_size; (void)ws_size;
}